// HotspotMSELoss_45226005627631
// MI455X (gfx1250) — compile-verified
//
#include <hip/hip_runtime.h>

typedef __attribute__((ext_vector_type(2))) float v2f;
typedef __attribute__((ext_vector_type(8))) float v8f;

#define BATCH   8
#define NPB     (1 << 21)      // 128*128*128 elements per batch
#define MASK_T  0.01f
#define QFRAC   0.9f
#define HOT_W   5.0f
#define EPS_F   1e-8f

#define PBLK    256            // blocks per batch for streaming kernels
#define TPB     256            // threads per block (8 wave32s)
#define VPT     8              // float4 loads per thread: PBLK*TPB*VPT*4 == NPB

// ---- workspace layout (u32 indices) ----
#define OFF_HIST1   0                             // [B][4096] u32
#define OFF_NVALID  (OFF_HIST1 + BATCH*4096)      // [B] u32
#define OFF_BIN1    (OFF_NVALID + 16)             // [B][2] u32
#define OFF_RREM1   (OFF_BIN1 + 16)               // [B][2] u32
#define OFF_BIN2    (OFF_RREM1 + 16)              // [B][2] u32
#define OFF_RREM2   (OFF_BIN2 + 16)               // [B][2] u32
#define OFF_FRAC    (OFF_RREM2 + 16)              // [B] f32
#define OFF_THRESH  (OFF_FRAC + 16)               // [B] f32
#define OFF_HIST2   (OFF_THRESH + 16)             // [B][2][4096] u32
#define OFF_HIST3   (OFF_HIST2 + BATCH*2*4096)    // [B][2][256] u32
#define OFF_PART    (OFF_HIST3 + BATCH*2*256)     // [B][PBLK][4] f32
#define OFF_SUMS    (OFF_PART + BATCH*PBLK*4)     // [B][4] f32
#define WS_ZERO_U32 (OFF_PART)                    // zero hist/select region only

// ---------------- zero scratch ----------------
__global__ void k_zero(unsigned* __restrict__ ws, int n) {
  int i = blockIdx.x * blockDim.x + threadIdx.x;
  if (i < n) ws[i] = 0u;
}

// ------------- level-1 histogram (top 12 bits of fp32 pattern) -------------
__global__ void k_hist1(const float* __restrict__ tgt, unsigned* __restrict__ ws) {
  __shared__ unsigned lh[4096];
  __shared__ unsigned lvalid;
  int b   = blockIdx.x / PBLK;
  int blk = blockIdx.x % PBLK;
  for (int i = threadIdx.x; i < 4096; i += TPB) lh[i] = 0u;
  if (threadIdx.x == 0) lvalid = 0u;
  __syncthreads();
  const float4* t = (const float4*)(tgt + (size_t)b * NPB + (size_t)blk * (TPB * VPT * 4));
  unsigned cnt = 0;
#pragma unroll
  for (int j = 0; j < VPT; ++j) {
    float4 v4 = t[j * TPB + threadIdx.x];
    float e[4] = {v4.x, v4.y, v4.z, v4.w};
#pragma unroll
    for (int c = 0; c < 4; ++c) {
      if (e[c] > MASK_T) {
        atomicAdd(&lh[__float_as_uint(e[c]) >> 20], 1u);
        ++cnt;
      }
    }
  }
  atomicAdd(&lvalid, cnt);
  __syncthreads();
  for (int i = threadIdx.x; i < 4096; i += TPB) {
    unsigned c = lh[i];
    if (c) atomicAdd(&ws[OFF_HIST1 + b * 4096 + i], c);
  }
  if (threadIdx.x == 0) atomicAdd(&ws[OFF_NVALID + b], lvalid);
}

// ------------- select level 1: find rank bins (mirrors torch.quantile pos math) -------------
__global__ void k_sel1(unsigned* __restrict__ ws) {
  int t = threadIdx.x;                 // one thread per (batch, rank)
  if (t >= BATCH * 2) return;
  int b = t >> 1, r = t & 1;
  unsigned n = ws[OFF_NVALID + b];
  float* wf = (float*)ws;
  unsigned bin = 0u, rrem = 0u;
  if (n > 0u) {
    // fp32 arithmetic exactly as the reference computes it
    float pos  = (float)(NPB - (int)n) + QFRAC * (float)((int)n - 1);
    float fpos = floorf(pos);
    int lo = (int)fpos;        if (lo < 0) lo = 0; if (lo > NPB - 1) lo = NPB - 1;
    int hi = (int)ceilf(pos);  if (hi < 0) hi = 0; if (hi > NPB - 1) hi = NPB - 1;
    if (r == 0) wf[OFF_FRAC + b] = pos - fpos;
    long rank = (long)(r ? hi : lo) - (long)(NPB - (long)n);
    if (rank < 0) rank = 0;
    if (rank >= (long)n) rank = (long)n - 1;
    unsigned rk = (unsigned)rank;
    unsigned cum = 0u;
    const unsigned* h = ws + OFF_HIST1 + b * 4096;
    for (int i = 0; i < 4096; ++i) {
      unsigned c = h[i];
      if (cum + c > rk) { bin = (unsigned)i; rrem = rk - cum; break; }
      cum += c;
    }
  } else {
    if (r == 0) wf[OFF_FRAC + b] = 0.f;
  }
  ws[OFF_BIN1 + t]  = bin;
  ws[OFF_RREM1 + t] = rrem;
}

// ------------- level-2 histogram (bits 19..8) within selected bins -------------
__global__ void k_hist2(const float* __restrict__ tgt, unsigned* __restrict__ ws) {
  __shared__ unsigned lh[2][4096];
  int b   = blockIdx.x / PBLK;
  int blk = blockIdx.x % PBLK;
  for (int i = threadIdx.x; i < 8192; i += TPB) ((unsigned*)lh)[i] = 0u;
  __syncthreads();
  unsigned b1lo = ws[OFF_BIN1 + b * 2 + 0];
  unsigned b1hi = ws[OFF_BIN1 + b * 2 + 1];
  const float4* t = (const float4*)(tgt + (size_t)b * NPB + (size_t)blk * (TPB * VPT * 4));
#pragma unroll
  for (int j = 0; j < VPT; ++j) {
    float4 v4 = t[j * TPB + threadIdx.x];
    float e[4] = {v4.x, v4.y, v4.z, v4.w};
#pragma unroll
    for (int c = 0; c < 4; ++c) {
      if (e[c] > MASK_T) {
        unsigned key = __float_as_uint(e[c]);
        unsigned top = key >> 20;
        unsigned mid = (key >> 8) & 0xFFFu;
        if (top == b1lo) atomicAdd(&lh[0][mid], 1u);
        if (top == b1hi) atomicAdd(&lh[1][mid], 1u);
      }
    }
  }
  __syncthreads();
  for (int i = threadIdx.x; i < 8192; i += TPB) {
    unsigned c = ((unsigned*)lh)[i];
    if (c) atomicAdd(&ws[OFF_HIST2 + b * 8192 + i], c);
  }
}

__global__ void k_sel2(unsigned* __restrict__ ws) {
  int t = threadIdx.x;
  if (t >= BATCH * 2) return;
  unsigned rk = ws[OFF_RREM1 + t];
  const unsigned* h = ws + OFF_HIST2 + t * 4096;
  unsigned bin = 0u, rrem = 0u, cum = 0u;
  for (int i = 0; i < 4096; ++i) {
    unsigned c = h[i];
    if (cum + c > rk) { bin = (unsigned)i; rrem = rk - cum; break; }
    cum += c;
  }
  ws[OFF_BIN2 + t]  = bin;
  ws[OFF_RREM2 + t] = rrem;
}

// ------------- level-3 histogram (bits 7..0) -------------
__global__ void k_hist3(const float* __restrict__ tgt, unsigned* __restrict__ ws) {
  __shared__ unsigned lh[2][256];
  int b   = blockIdx.x / PBLK;
  int blk = blockIdx.x % PBLK;
  for (int i = threadIdx.x; i < 512; i += TPB) ((unsigned*)lh)[i] = 0u;
  __syncthreads();
  unsigned plo = (ws[OFF_BIN1 + b * 2 + 0] << 12) | ws[OFF_BIN2 + b * 2 + 0];
  unsigned phi = (ws[OFF_BIN1 + b * 2 + 1] << 12) | ws[OFF_BIN2 + b * 2 + 1];
  const float4* t = (const float4*)(tgt + (size_t)b * NPB + (size_t)blk * (TPB * VPT * 4));
#pragma unroll
  for (int j = 0; j < VPT; ++j) {
    float4 v4 = t[j * TPB + threadIdx.x];
    float e[4] = {v4.x, v4.y, v4.z, v4.w};
#pragma unroll
    for (int c = 0; c < 4; ++c) {
      if (e[c] > MASK_T) {
        unsigned key = __float_as_uint(e[c]);
        unsigned pfx = key >> 8;
        if (pfx == plo) atomicAdd(&lh[0][key & 0xFFu], 1u);
        if (pfx == phi) atomicAdd(&lh[1][key & 0xFFu], 1u);
      }
    }
  }
  __syncthreads();
  for (int i = threadIdx.x; i < 512; i += TPB) {
    unsigned c = ((unsigned*)lh)[i];
    if (c) atomicAdd(&ws[OFF_HIST3 + b * 512 + i], c);
  }
}

// ------------- final select + threshold interpolation -------------
__global__ void k_sel3(unsigned* __restrict__ ws) {
  __shared__ float vsh[BATCH * 2];
  int t = threadIdx.x;
  if (t < BATCH * 2) {
    unsigned rk = ws[OFF_RREM2 + t];
    const unsigned* h = ws + OFF_HIST3 + t * 256;
    unsigned bin = 0u, cum = 0u;
    for (int i = 0; i < 256; ++i) {
      unsigned c = h[i];
      if (cum + c > rk) { bin = (unsigned)i; break; }
      cum += c;
    }
    unsigned key = (ws[OFF_BIN1 + t] << 20) | (ws[OFF_BIN2 + t] << 8) | bin;
    vsh[t] = __uint_as_float(key);
  }
  __syncthreads();
  if (t < BATCH) {
    float* wf = (float*)ws;
    unsigned n = ws[OFF_NVALID + t];
    float thr;
    if (n > 0u) {
      float vlo = vsh[t * 2 + 0], vhi = vsh[t * 2 + 1];
      thr = vlo + wf[OFF_FRAC + t] * (vhi - vlo);
    } else {
      thr = MASK_T;
    }
    wf[OFF_THRESH + t] = thr;
  }
}

// ------------- fused masked-MSE pass: per-block partial sums (deterministic) -------------
__global__ void k_loss(const float* __restrict__ prd, const float* __restrict__ tgt,
                       unsigned* __restrict__ ws) {
  __shared__ float wsum[TPB / 32][4];
  int b   = blockIdx.x / PBLK;
  int blk = blockIdx.x % PBLK;
  float thr = ((const float*)ws)[OFF_THRESH + b];
  size_t base = (size_t)b * NPB + (size_t)blk * (TPB * VPT * 4);
  const float4* t4 = (const float4*)(tgt + base);
  const float4* p4 = (const float4*)(prd + base);
  float sh = 0.f, sb = 0.f, nh = 0.f, nb = 0.f;
#pragma unroll
  for (int j = 0; j < VPT; ++j) {
    float4 tv4 = t4[j * TPB + threadIdx.x];
    float4 pv4 = p4[j * TPB + threadIdx.x];
    float te[4] = {tv4.x, tv4.y, tv4.z, tv4.w};
    float pe[4] = {pv4.x, pv4.y, pv4.z, pv4.w};
#pragma unroll
    for (int c = 0; c < 4; ++c) {
      float tv = te[c], pv = pe[c];
      bool  m  = tv > MASK_T;
      float mt = m ? tv : 0.f;
      float mp = m ? pv : 0.f;
      float e  = mp - mt;
      float e2 = e * e;
      bool hot = mt > thr;           // hot implies m (thr > MASK_T > 0)
      bool bg  = m && !hot;
      sh += hot ? e2 : 0.f;  nh += hot ? 1.f : 0.f;
      sb += bg  ? e2 : 0.f;  nb += bg  ? 1.f : 0.f;
    }
  }
  // wave32 butterfly
  for (int off = 16; off >= 1; off >>= 1) {
    sh += __shfl_xor(sh, off, 32);
    sb += __shfl_xor(sb, off, 32);
    nh += __shfl_xor(nh, off, 32);
    nb += __shfl_xor(nb, off, 32);
  }
  int wave = threadIdx.x >> 5, lane = threadIdx.x & 31;
  if (lane == 0) { wsum[wave][0] = sh; wsum[wave][1] = sb; wsum[wave][2] = nh; wsum[wave][3] = nb; }
  __syncthreads();
  if (threadIdx.x == 0) {
    float a0 = 0.f, a1 = 0.f, a2 = 0.f, a3 = 0.f;
    for (int w = 0; w < TPB / 32; ++w) {
      a0 += wsum[w][0]; a1 += wsum[w][1]; a2 += wsum[w][2]; a3 += wsum[w][3];
    }
    float4* out = (float4*)((float*)ws + OFF_PART + (size_t)(b * PBLK + blk) * 4);
    *out = make_float4(a0, a1, a2, a3);
  }
}

// ------------- per-batch reduction of block partials via V_WMMA_F32_16X16X4_F32 -------------
// A = 64 partials striped over the wave, B = ones(4x16), C chained accumulator.
// D[m][n] = row-sum(A row m) for every n; summing the 8 C VGPRs per lane and
// folding lane<->lane+16 yields the exact fp32 total (x*1.0 accumulated in f32 C).
__global__ void k_reduce(unsigned* __restrict__ ws) {
  int b    = blockIdx.x;
  int lane = threadIdx.x;          // single wave32, EXEC all ones
  const float* part = (const float*)ws + OFF_PART + (size_t)b * PBLK * 4;
  float* sums = (float*)ws + OFF_SUMS + b * 4;
  v2f ones; ones.x = 1.f; ones.y = 1.f;
#pragma unroll
  for (int q = 0; q < 4; ++q) {
    v8f acc = {};
#pragma unroll
    for (int c = 0; c < PBLK / 64; ++c) {   // 4 chunks of 64 partials
      int base = c * 64 + lane * 2;
      v2f a;
      a.x = part[(base + 0) * 4 + q];
      a.y = part[(base + 1) * 4 + q];
      acc = __builtin_amdgcn_wmma_f32_16x16x4_f32(
          /*neg_a=*/false, a, /*neg_b=*/false, ones,
          /*c_mod=*/(short)0, acc, /*reuse_a=*/false, /*reuse_b=*/false);
    }
    float s = acc[0] + acc[1] + acc[2] + acc[3] + acc[4] + acc[5] + acc[6] + acc[7];
    s += __shfl_xor(s, 16, 32);
    if (lane == 0) sums[q] = s;
  }
}

// ------------- combine: 5*hot_loss + bg_loss, mean over batches -------------
__global__ void k_final(const unsigned* __restrict__ ws, float* __restrict__ out) {
  if (blockIdx.x == 0 && threadIdx.x == 0) {
    const float* s = (const float*)ws + OFF_SUMS;
    float accum = 0.f;
    for (int b = 0; b < BATCH; ++b) {
      float sh = s[b * 4 + 0], sb = s[b * 4 + 1];
      float nh = s[b * 4 + 2], nb = s[b * 4 + 3];
      accum += HOT_W * (sh / (nh + EPS_F)) + sb / (nb + EPS_F);
    }
    out[0] = accum / (float)BATCH;
  }
}

extern "C" void kernel_launch(void* const* d_in, const int* in_sizes, int n_in,
                              void* d_out, int out_size, void* d_ws, size_t ws_size,
                              hipStream_t stream) {
  const float* pred = (const float*)d_in[0];
  const float* tgt  = (const float*)d_in[1];
  unsigned* ws = (unsigned*)d_ws;
  float* out = (float*)d_out;
  (void)in_sizes; (void)n_in; (void)out_size; (void)ws_size;

  int zn = WS_ZERO_U32;
  k_zero <<<(zn + 255) / 256, 256, 0, stream>>>(ws, zn);
  k_hist1<<<BATCH * PBLK, TPB, 0, stream>>>(tgt, ws);
  k_sel1 <<<1, 32, 0, stream>>>(ws);
  k_hist2<<<BATCH * PBLK, TPB, 0, stream>>>(tgt, ws);
  k_sel2 <<<1, 32, 0, stream>>>(ws);
  k_hist3<<<BATCH * PBLK, TPB, 0, stream>>>(tgt, ws);
  k_sel3 <<<1, 32, 0, stream>>>(ws);
  k_loss <<<BATCH * PBLK, TPB, 0, stream>>>(pred, tgt, ws);
  k_reduce<<<BATCH, 32, 0, stream>>>(ws);
  k_final<<<1, 32, 0, stream>>>(ws, out);
}